// MutagGCN_87041807221071
// MI455X (gfx1250) — compile-verified
//
#include <hip/hip_runtime.h>
#include <hip/hip_bf16.h>

// ---------------------------------------------------------------------------
// MutagGCN for MI455X (gfx1250, wave32).
// Node MLPs use V_WMMA_F32_16X16X4_F32 (exact fp32). Edge aggregation uses
// global f32 atomics (L2-resident working set, 192MB L2). Pooling uses the
// monotone-bits trick (relu outputs >= 0) with unsigned atomicMax.
// ---------------------------------------------------------------------------

typedef __attribute__((ext_vector_type(2))) float v2f;
typedef __attribute__((ext_vector_type(8))) float v8f;

static __device__ __forceinline__ v8f wmma_f32(v2f a, v2f b, v8f c) {
  // (neg_a, A, neg_b, B, c_mod, C, reuse_a, reuse_b)
  return __builtin_amdgcn_wmma_f32_16x16x4_f32(false, a, false, b, (short)0, c,
                                               false, false);
}

__global__ __launch_bounds__(256) void zero_kernel(float* __restrict__ p, long n) {
  long i = (long)blockIdx.x * blockDim.x + threadIdx.x;
  long stride = (long)gridDim.x * blockDim.x;
  for (; i < n; i += stride) p[i] = 0.0f;
}

// Layer-1 edge scatter: msg = [edge_feat(4), node_feat[src](7)] summed into
// acc1[dst][0..10] (stride 12, slot 11 stays 0 as K-padding); deg counts edges.
__global__ __launch_bounds__(256) void scatter1_kernel(
    const float* __restrict__ node_feat, const float* __restrict__ edge_feat,
    const int* __restrict__ src, const int* __restrict__ dst,
    float* __restrict__ acc1, float* __restrict__ deg, int E) {
  int e = blockIdx.x * 256 + threadIdx.x;
  if (e >= E) return;
  int s = src[e], d = dst[e];
  const float4 ef = reinterpret_cast<const float4*>(edge_feat)[e];
  float* a = acc1 + (long)d * 12;
  atomicAdd(a + 0, ef.x);
  atomicAdd(a + 1, ef.y);
  atomicAdd(a + 2, ef.z);
  atomicAdd(a + 3, ef.w);
  const float* nf = node_feat + (long)s * 7;
#pragma unroll
  for (int i = 0; i < 7; ++i) atomicAdd(a + 4 + i, nf[i]);
  atomicAdd(deg + d, 1.0f);
}

// Layer-2 edge scatter: acc2[dst] += h1[src] (32 feats). 8 threads/edge,
// float4 per thread for coalesced gathers and fewer atomics per thread.
__global__ __launch_bounds__(256) void scatter2_kernel(
    const float* __restrict__ h1, const int* __restrict__ src,
    const int* __restrict__ dst, float* __restrict__ acc2, int E) {
  long gid = (long)blockIdx.x * 256 + threadIdx.x;
  if (gid >= (long)E * 8) return;
  int e = (int)(gid >> 3);
  int c = (int)(gid & 7);
  int s = src[e], d = dst[e];
  const float4 v = reinterpret_cast<const float4*>(h1 + (long)s * 32)[c];
  float* a = acc2 + (long)d * 32 + c * 4;
  atomicAdd(a + 0, v.x);
  atomicAdd(a + 1, v.y);
  atomicAdd(a + 2, v.z);
  atomicAdd(a + 3, v.w);
}

// Node MLP: h = relu(relu((acc/max(deg,1)) @ Wa + ba) @ Wb + bb).
// One wave per 16-node tile; two 16x16 fp32 WMMA accumulators cover N=32.
// KIN = padded input width (A stride), KVALID = real rows of Wa.
template <int KIN, int KVALID>
__global__ __launch_bounds__(256) void mlp_kernel(
    const float* __restrict__ acc, const float* __restrict__ deg,
    const float* __restrict__ Wa, const float* __restrict__ ba,
    const float* __restrict__ Wb, const float* __restrict__ bb,
    float* __restrict__ out, int n_nodes) {
  __shared__ float smid[8][16 * 33];  // per-wave 16x32 tile, stride 33 (bank-safe)
  const int lane = threadIdx.x & 31;
  const int wave = threadIdx.x >> 5;
  const int tile = blockIdx.x * 8 + wave;
  const int ntiles = (n_nodes + 15) >> 4;
  if (tile >= ntiles) return;  // wave-uniform: EXEC stays all-ones for WMMA

  const int m = lane & 15;   // A row / C row-group base
  const int kh = lane >> 4;  // K half-select for A/B frags
  const int n0 = m;          // B/C column for this lane
  const int node = tile * 16 + m;
  const int node_c = node < n_nodes ? node : (n_nodes - 1);
  const float inv = 1.0f / fmaxf(deg[node_c], 1.0f);

  // ---- GEMM1: [16 x KIN] @ [KVALID x 32] ----
  v8f c0 = {}, c1 = {};
#pragma unroll
  for (int ks = 0; ks < KIN / 4; ++ks) {
    const int k = ks * 4 + kh * 2;
    v2f a, b0, b1;
    a.x = acc[(long)node_c * KIN + k] * inv;
    a.y = acc[(long)node_c * KIN + k + 1] * inv;  // pad slot is zeroed
    b0.x = (k < KVALID) ? Wa[(k)*32 + n0] : 0.0f;
    b0.y = (k + 1 < KVALID) ? Wa[(k + 1) * 32 + n0] : 0.0f;
    b1.x = (k < KVALID) ? Wa[(k)*32 + n0 + 16] : 0.0f;
    b1.y = (k + 1 < KVALID) ? Wa[(k + 1) * 32 + n0 + 16] : 0.0f;
    c0 = wmma_f32(a, b0, c0);
    c1 = wmma_f32(a, b1, c1);
  }
  {
    const float bias0 = ba[n0], bias1 = ba[n0 + 16];
#pragma unroll
    for (int j = 0; j < 8; ++j) {
      const int mm = j + 8 * kh;  // C/D row = VGPR j + 8*(lane>=16)
      smid[wave][mm * 33 + n0] = fmaxf(c0[j] + bias0, 0.0f);
      smid[wave][mm * 33 + n0 + 16] = fmaxf(c1[j] + bias1, 0.0f);
    }
  }

  // ---- GEMM2: [16 x 32] @ [32 x 32] (A from per-wave LDS tile) ----
  v8f d0 = {}, d1 = {};
#pragma unroll
  for (int ks = 0; ks < 8; ++ks) {
    const int k = ks * 4 + kh * 2;
    v2f a, b0, b1;
    a.x = smid[wave][m * 33 + k];
    a.y = smid[wave][m * 33 + k + 1];
    b0.x = Wb[(k)*32 + n0];
    b0.y = Wb[(k + 1) * 32 + n0];
    b1.x = Wb[(k)*32 + n0 + 16];
    b1.y = Wb[(k + 1) * 32 + n0 + 16];
    d0 = wmma_f32(a, b0, d0);
    d1 = wmma_f32(a, b1, d1);
  }
  {
    const float bias0 = bb[n0], bias1 = bb[n0 + 16];
#pragma unroll
    for (int j = 0; j < 8; ++j) {
      const int mm = j + 8 * kh;
      const int on = tile * 16 + mm;
      if (on < n_nodes) {
        out[(long)on * 32 + n0] = fmaxf(d0[j] + bias0, 0.0f);
        out[(long)on * 32 + n0 + 16] = fmaxf(d1[j] + bias1, 0.0f);
      }
    }
  }
}

// Per-graph max pool. h2 >= 0 (relu) so fp32 bit patterns order like floats
// as unsigned ints; pooled is zero-initialized (== 0.0f, a valid lower bound).
__global__ __launch_bounds__(256) void pool_kernel(
    const float* __restrict__ h2, const int* __restrict__ graph_ids,
    unsigned int* __restrict__ pooled_u, int n_nodes) {
  long gid = (long)blockIdx.x * 256 + threadIdx.x;
  if (gid >= (long)n_nodes * 32) return;
  int node = (int)(gid >> 5);
  int f = (int)(gid & 31);
  int g = graph_ids[node];
  atomicMax(pooled_u + (long)g * 32 + f, __float_as_uint(h2[gid]));
}

// Tiny classifier + 2-class softmax: one thread per graph.
__global__ __launch_bounds__(64) void classifier_kernel(
    const float* __restrict__ pooled, const float* __restrict__ Wm1,
    const float* __restrict__ bm1, const float* __restrict__ Wm2,
    const float* __restrict__ bm2, float* __restrict__ out, int n_graphs) {
  int g = blockIdx.x * blockDim.x + threadIdx.x;
  if (g >= n_graphs) return;
  float p[32];
#pragma unroll
  for (int k = 0; k < 32; ++k) p[k] = pooled[(long)g * 32 + k];
  float l0 = bm2[0], l1 = bm2[1];
#pragma unroll
  for (int j = 0; j < 16; ++j) {
    float t = bm1[j];
#pragma unroll
    for (int k = 0; k < 32; ++k) t += p[k] * Wm1[k * 16 + j];
    t = fmaxf(t, 0.0f);
    l0 += t * Wm2[j * 2 + 0];
    l1 += t * Wm2[j * 2 + 1];
  }
  float mx = fmaxf(l0, l1);
  float e0 = expf(l0 - mx), e1 = expf(l1 - mx);
  float s = e0 + e1;
  out[g * 2 + 0] = e0 / s;
  out[g * 2 + 1] = e1 / s;
}

extern "C" void kernel_launch(void* const* d_in, const int* in_sizes, int n_in,
                              void* d_out, int out_size, void* d_ws,
                              size_t ws_size, hipStream_t stream) {
  const float* node_feat = (const float*)d_in[0];
  const float* edge_feat = (const float*)d_in[1];
  const int* src = (const int*)d_in[2];
  const int* dst = (const int*)d_in[3];
  const int* graph_ids = (const int*)d_in[4];
  const float* W1a = (const float*)d_in[5];
  const float* b1a = (const float*)d_in[6];
  const float* W1b = (const float*)d_in[7];
  const float* b1b = (const float*)d_in[8];
  const float* W2a = (const float*)d_in[9];
  const float* b2a = (const float*)d_in[10];
  const float* W2b = (const float*)d_in[11];
  const float* b2b = (const float*)d_in[12];
  const float* Wm1 = (const float*)d_in[13];
  const float* bm1 = (const float*)d_in[14];
  const float* Wm2 = (const float*)d_in[15];
  const float* bm2 = (const float*)d_in[16];

  const int n_nodes = in_sizes[0] / 7;
  const int E = in_sizes[3];
  const int n_graphs = out_size / 2;

  // Workspace layout (floats); zeroed region [acc1 .. pooled] is contiguous.
  float* ws = (float*)d_ws;
  long off = 0;
  float* acc1 = ws + off;   off += (long)n_nodes * 12;          // padded to K=12
  float* deg = ws + off;    off += n_nodes;                     // shared by both layers
  float* acc2 = ws + off;   off += (long)n_nodes * 32;
  float* pooled = ws + off; off += (((long)n_graphs * 32 + 15) / 16) * 16;
  const long zero_n = off;
  float* h1 = ws + off;     off += (long)n_nodes * 32;
  float* h2 = ws + off;     off += (long)n_nodes * 32;
  (void)ws_size; (void)n_in;  // ~88MB required

  zero_kernel<<<(int)((zero_n + 255) / 256), 256, 0, stream>>>(acc1, zero_n);

  scatter1_kernel<<<(E + 255) / 256, 256, 0, stream>>>(node_feat, edge_feat,
                                                       src, dst, acc1, deg, E);

  const int ntiles = (n_nodes + 15) / 16;
  mlp_kernel<12, 11><<<(ntiles + 7) / 8, 256, 0, stream>>>(acc1, deg, W1a, b1a,
                                                           W1b, b1b, h1, n_nodes);

  const long t2 = (long)E * 8;
  scatter2_kernel<<<(int)((t2 + 255) / 256), 256, 0, stream>>>(h1, src, dst,
                                                               acc2, E);

  mlp_kernel<32, 32><<<(ntiles + 7) / 8, 256, 0, stream>>>(acc2, deg, W2a, b2a,
                                                           W2b, b2b, h2, n_nodes);

  const long t3 = (long)n_nodes * 32;
  pool_kernel<<<(int)((t3 + 255) / 256), 256, 0, stream>>>(
      h2, graph_ids, (unsigned int*)pooled, n_nodes);

  classifier_kernel<<<(n_graphs + 63) / 64, 64, 0, stream>>>(
      pooled, Wm1, bm1, Wm2, bm2, (float*)d_out, n_graphs);
}